// HresModel_65678639891305
// MI455X (gfx1250) — compile-verified
//
#include <hip/hip_runtime.h>
#include <math.h>

// ---------------------------------------------------------------------------
// MI455X (gfx1250, wave32) implementation of the HRES 2-stage NAT model.
// fp32 end-to-end; all large GEMMs run through V_WMMA_F32_16X16X4_F32.
// ---------------------------------------------------------------------------

typedef float v2f __attribute__((ext_vector_type(2)));
typedef float v8f __attribute__((ext_vector_type(8)));

#define ACT_NONE 0
#define ACT_RELU 1
#define ACT_GELU 2

// ---------------------------------------------------------------------------
// Generic WMMA GEMM:  Y[M,N] = act( X[M,K] @ W + bias + addsrc )
//   W element (k,n) at  W[k*wks + n*wns]   (wns!=1 used for the patchify conv)
//   scatter=1 remaps output row m -> pixel (m/Wp*f+fy, m%Wp*f+fx) of a
//   (.. x wOut x N) HWC tensor (the conv "bohfwg" einsum).
// Block: 256 thr = 8 waves as 2(M) x 4(N); block tile 32x64; K step 16.
// Requires M%32==0, N%64==0, K%16==0 (true for every GEMM routed here).
// ---------------------------------------------------------------------------
__global__ __launch_bounds__(256) void k_gemm_wmma(
    const float* __restrict__ X, const float* __restrict__ W,
    const float* __restrict__ bias, const float* __restrict__ addsrc,
    float* __restrict__ Y, int M, int N, int K, long wks, long wns, int act,
    int scatter, int Wp, int f, int fy, int fx, int wOut) {
  __shared__ float As[16][33];  // [k][m]
  __shared__ float Bs[16][65];  // [k][n]

  const int tid  = threadIdx.x;
  const int lane = tid & 31;
  const int wave = tid >> 5;
  const int wm   = wave >> 2;  // 0..1
  const int wn   = wave & 3;   // 0..3
  const int bm   = blockIdx.y * 32;
  const int bn   = blockIdx.x * 64;

  // A-tile cooperative load mapping: one float2 per thread
  const int a_m = tid >> 3;         // 0..31
  const int a_k = (tid & 7) << 1;   // 0,2,..,14
  // B-tile cooperative load mapping: four floats per thread
  const int b_k  = (tid << 2) >> 6;  // 0..15
  const int b_n0 = (tid << 2) & 63;

  // wmma fragment lane mapping (16x16x4 f32)
  const int lm   = lane & 15;
  const int koff = (lane >> 4) << 1;  // 0 or 2
  const int mrow = wm * 16 + lm;
  const int ncol = wn * 16 + lm;

  v8f acc = {};

  for (int kk = 0; kk < K; kk += 16) {
    {
      const float* src = X + (size_t)(bm + a_m) * K + kk + a_k;
      float2 v = *(const float2*)src;
      As[a_k][a_m]     = v.x;
      As[a_k + 1][a_m] = v.y;
      if (kk + 16 < K)  // emits global_prefetch_b8 for the next K tile
        __builtin_prefetch(src + 16, 0, 1);
    }
    {
      const float* wp = W + (size_t)(kk + b_k) * wks + (size_t)(bn + b_n0) * wns;
#pragma unroll
      for (int j = 0; j < 4; ++j) Bs[b_k][b_n0 + j] = wp[(size_t)j * wns];
    }
    __syncthreads();
#pragma unroll
    for (int ks = 0; ks < 16; ks += 4) {
      v2f a, b;
      a.x = As[ks + koff][mrow];
      a.y = As[ks + koff + 1][mrow];
      b.x = Bs[ks + koff][ncol];
      b.y = Bs[ks + koff + 1][ncol];
      acc = __builtin_amdgcn_wmma_f32_16x16x4_f32(false, a, false, b, (short)0,
                                                  acc, false, false);
    }
    __syncthreads();
  }

  const int gn    = bn + ncol;
  const int mbase = bm + wm * 16 + ((lane >> 4) << 3);
#pragma unroll
  for (int i = 0; i < 8; ++i) {
    int gm = mbase + i;
    size_t idx;
    if (scatter) {
      int yy = (gm / Wp) * f + fy;
      int xx = (gm % Wp) * f + fx;
      idx = ((size_t)yy * wOut + xx) * N + gn;
    } else {
      idx = (size_t)gm * N + gn;
    }
    float v = acc[i];
    if (bias) v += bias[gn];
    if (addsrc) v += addsrc[idx];
    if (act == ACT_RELU)
      v = fmaxf(v, 0.f);
    else if (act == ACT_GELU)
      v = 0.5f * v * (1.f + erff(v * 0.70710678118654752f));
    Y[idx] = v;
  }
}

// ---------------------------------------------------------------------------
// LayerNorm over last dim C, one wave32 per token (coalesced, shfl reduce)
// ---------------------------------------------------------------------------
__global__ __launch_bounds__(256) void k_layernorm(
    const float* __restrict__ x, const float* __restrict__ g,
    const float* __restrict__ b, float* __restrict__ y, int ntok, int C) {
  int gid  = blockIdx.x * blockDim.x + threadIdx.x;
  int wid  = gid >> 5;
  int lane = gid & 31;
  if (wid >= ntok) return;
  const float* row = x + (size_t)wid * C;
  float s = 0.f;
  for (int c = lane; c < C; c += 32) s += row[c];
#pragma unroll
  for (int o = 16; o > 0; o >>= 1) s += __shfl_xor(s, o, 32);
  float mu = s / C;
  float v = 0.f;
  for (int c = lane; c < C; c += 32) {
    float d = row[c] - mu;
    v += d * d;
  }
#pragma unroll
  for (int o = 16; o > 0; o >>= 1) v += __shfl_xor(v, o, 32);
  float rs = rsqrtf(v / C + 1e-5f);
  float* orow = y + (size_t)wid * C;
  for (int c = lane; c < C; c += 32) orow[c] = (row[c] - mu) * rs * g[c] + b[c];
}

// ---------------------------------------------------------------------------
// 7x7 neighborhood attention, head_dim=16, one thread per (pixel, head).
// qkv row layout: [3][nh][16]; writes a[pixel, head*16+d].
// The whole qkv tensor (<=151MB) is ~L2-resident on MI455X (192MB L2).
// ---------------------------------------------------------------------------
__device__ __forceinline__ float dot4(float4 a, float4 b) {
  return a.x * b.x + a.y * b.y + a.z * b.z + a.w * b.w;
}

__global__ __launch_bounds__(256) void k_na2d(const float* __restrict__ qkv,
                                              float* __restrict__ outA, int H,
                                              int W, int nh) {
  int gid = blockIdx.x * blockDim.x + threadIdx.x;
  int total = H * W * nh;
  if (gid >= total) return;
  int head = gid % nh;
  int pix  = gid / nh;
  int y    = pix / W;
  int x    = pix % W;
  int C3   = 3 * nh * 16;

  const float4* qp = (const float4*)(qkv + (size_t)pix * C3 + head * 16);
  float4 Q0 = qp[0], Q1 = qp[1], Q2 = qp[2], Q3 = qp[3];

  int sh = min(max(y - 3, 0), H - 7);
  int sw = min(max(x - 3, 0), W - 7);

  float s[49];
  float mx = -1e30f;
#pragma unroll
  for (int a = 0; a < 7; ++a) {
#pragma unroll
    for (int b = 0; b < 7; ++b) {
      const float4* kp = (const float4*)(qkv +
          ((size_t)(sh + a) * W + (sw + b)) * C3 + (nh + head) * 16);
      float d = dot4(Q0, kp[0]) + dot4(Q1, kp[1]) + dot4(Q2, kp[2]) +
                dot4(Q3, kp[3]);
      d *= 0.25f;  // hd^-0.5 = 16^-0.5
      s[a * 7 + b] = d;
      mx = fmaxf(mx, d);
    }
  }
  float sum = 0.f;
#pragma unroll
  for (int i = 0; i < 49; ++i) {
    s[i] = expf(s[i] - mx);
    sum += s[i];
  }
  float inv = 1.f / sum;
  float o[16];
#pragma unroll
  for (int d = 0; d < 16; ++d) o[d] = 0.f;
#pragma unroll
  for (int a = 0; a < 7; ++a) {
#pragma unroll
    for (int b = 0; b < 7; ++b) {
      float wgt = s[a * 7 + b] * inv;
      const float4* vp = (const float4*)(qkv +
          ((size_t)(sh + a) * W + (sw + b)) * C3 + (2 * nh + head) * 16);
#pragma unroll
      for (int j = 0; j < 4; ++j) {
        float4 V = vp[j];
        o[4 * j + 0] += wgt * V.x;
        o[4 * j + 1] += wgt * V.y;
        o[4 * j + 2] += wgt * V.z;
        o[4 * j + 3] += wgt * V.w;
      }
    }
  }
  float* op = outA + (size_t)pix * (nh * 16) + head * 16;
#pragma unroll
  for (int d = 0; d < 16; ++d) op[d] = o[d];
}

// ---------------------------------------------------------------------------
// Per-step encode: writes  out[y,x,o] = conv_b[o] + enc(pooled)[o] + posemb
// pooled = [elev-meanpool(4) | onehot17(modis-maxpool) | bilinear16(x0)]
// ---------------------------------------------------------------------------
__global__ __launch_bounds__(256) void k_encode_pe(
    const float* __restrict__ elev, const int* __restrict__ modis,
    const float* __restrict__ x0, const float* __restrict__ enc_w,
    const float* __restrict__ enc_b, const float* __restrict__ conv_b,
    float* __restrict__ out, int h, int w, int L, int ff) {
  int gid = blockIdx.x * blockDim.x + threadIdx.x;
  if (gid >= h * w) return;
  int y = gid / w, x = gid % w;

  float pooled[37];
  float e0 = 0.f, e1 = 0.f, e2 = 0.f, e3 = 0.f;
  int mmax = 0;
  for (int dy = 0; dy < ff; ++dy)
    for (int dx = 0; dx < ff; ++dx) {
      int Y = y * ff + dy, X = x * ff + dx;
      const float* ep = elev + ((size_t)Y * 256 + X) * 4;
      e0 += ep[0]; e1 += ep[1]; e2 += ep[2]; e3 += ep[3];
      int m = modis[Y * 256 + X];
      mmax = max(mmax, m);
    }
  float inv = 1.f / (float)(ff * ff);
  pooled[0] = e0 * inv; pooled[1] = e1 * inv;
  pooled[2] = e2 * inv; pooled[3] = e3 * inv;
#pragma unroll
  for (int i = 0; i < 17; ++i) pooled[4 + i] = (i == mmax) ? 1.f : 0.f;

  // bilinear (align-corners) upsample of 32x32x16 initial concat
  float ry = (float)y * (31.f / (float)(h - 1));
  float rx = (float)x * (31.f / (float)(w - 1));
  int y0 = (int)floorf(ry), x0i = (int)floorf(rx);
  int y1 = min(y0 + 1, 31), x1 = min(x0i + 1, 31);
  float wy = ry - (float)y0, wx = rx - (float)x0i;
  const float* p00 = x0 + ((size_t)y0 * 32 + x0i) * 16;
  const float* p01 = x0 + ((size_t)y0 * 32 + x1) * 16;
  const float* p10 = x0 + ((size_t)y1 * 32 + x0i) * 16;
  const float* p11 = x0 + ((size_t)y1 * 32 + x1) * 16;
#pragma unroll
  for (int c = 0; c < 16; ++c) {
    float top = p00[c] * (1.f - wx) + p01[c] * wx;
    float bot = p10[c] * (1.f - wx) + p11[c] * wx;
    pooled[21 + c] = top * (1.f - wy) + bot * wy;
  }

  int Lq = L / 4;
  float invq = 1.f / (float)(Lq - 1);
  float* orow = out + (size_t)gid * L;
  for (int o = 0; o < L; ++o) {
    int sec = o / Lq;
    int j   = o - sec * Lq;
    float om  = powf(10000.f, -(float)j * invq);
    float ang = ((sec < 2) ? (float)x : (float)y) * om;
    float pe  = ((sec & 1) == 0) ? sinf(ang) : cosf(ang);
    float acc = enc_b[o] + conv_b[o] + pe;
    const float* wr = enc_w + (size_t)o * 37;
#pragma unroll
    for (int c = 0; c < 37; ++c) acc += wr[c] * pooled[c];
    orow[o] = acc;
  }
}

// ---------------------------------------------------------------------------
// Concat inputs into x0 (32x32x16 HWC)
// ---------------------------------------------------------------------------
__global__ void k_concat(const float* __restrict__ era5,
                         const float* __restrict__ rad,
                         const float* __restrict__ ang,
                         const float* __restrict__ st,
                         float* __restrict__ out) {
  int pix = blockIdx.x * blockDim.x + threadIdx.x;
  if (pix >= 1024) return;
  float* o = out + (size_t)pix * 16;
#pragma unroll
  for (int c = 0; c < 8; ++c) o[c] = era5[pix * 8 + c];
  o[8]  = rad[pix];
  o[9]  = ang[pix * 2];
  o[10] = ang[pix * 2 + 1];
#pragma unroll
  for (int c = 0; c < 5; ++c) o[11 + c] = st[pix * 5 + c];
}

// ---------------------------------------------------------------------------
// Final head: out[t,n] = hidden[t,:384] @ out_w2[:,n] + b2[n]   (N=8 < WMMA)
// ---------------------------------------------------------------------------
__global__ __launch_bounds__(256) void k_head2(const float* __restrict__ hid,
                                               const float* __restrict__ w2,
                                               const float* __restrict__ b2,
                                               float* __restrict__ out,
                                               int ntok) {
  int gid = blockIdx.x * blockDim.x + threadIdx.x;
  if (gid >= ntok * 8) return;
  int t = gid >> 3, n = gid & 7;
  const float* hr = hid + (size_t)t * 384;
  float acc = b2[n];
  for (int k = 0; k < 384; ++k) acc += hr[k] * w2[k * 8 + n];
  out[gid] = acc;
}

// ---------------------------------------------------------------------------
// Host orchestration
// ---------------------------------------------------------------------------
extern "C" void kernel_launch(void* const* d_in, const int* in_sizes, int n_in,
                              void* d_out, int out_size, void* d_ws,
                              size_t ws_size, hipStream_t stream) {
  (void)in_sizes; (void)n_in; (void)out_size; (void)ws_size;
  const float* era5  = (const float*)d_in[0];
  const float* rad   = (const float*)d_in[1];
  const float* ang   = (const float*)d_in[2];
  const float* stat  = (const float*)d_in[3];
  const float* elev  = (const float*)d_in[4];
  const int*   modis = (const int*)d_in[5];

  // params flattened in jax tree_leaves order (sorted dict keys)
  int p = 6;
  auto F = [&](int i) { return (const float*)d_in[i]; };
  const float* out_b1 = F(p++);
  const float* out_b2 = F(p++);
  const float* out_w1 = F(p++);
  const float* out_w2 = F(p++);
  struct Blk {
    const float *fc1_b, *fc1_w, *fc2_b, *fc2_w, *n1_b, *n1_g, *n2_b, *n2_g,
        *proj_b, *proj_w, *qkv_b, *qkv_w;
  };
  struct Step {
    Blk blk[2];
    const float *conv_b, *conv_w, *enc_b, *enc_w;
  };
  Step st2[2];
  for (int s = 0; s < 2; ++s) {
    for (int bi = 0; bi < 2; ++bi) {
      Blk& B = st2[s].blk[bi];
      B.fc1_b = F(p++); B.fc1_w = F(p++); B.fc2_b = F(p++); B.fc2_w = F(p++);
      B.n1_b = F(p++);  B.n1_g = F(p++);  B.n2_b = F(p++);  B.n2_g = F(p++);
      B.proj_b = F(p++); B.proj_w = F(p++); B.qkv_b = F(p++); B.qkv_w = F(p++);
    }
    st2[s].conv_b = F(p++); st2[s].conv_w = F(p++);
    st2[s].enc_b  = F(p++); st2[s].enc_w  = F(p++);
  }

  // workspace (floats): x0 | R1 | R2 | R3(big)  -> ~302 MB total
  float* ws  = (float*)d_ws;
  float* x0b = ws;                       // 16384
  float* R1  = x0b + 16384;              // 256*256*192 = 12,582,912
  float* R2  = R1 + 12582912;            // 12,582,912
  float* R3  = R2 + 12582912;            // up to 256*256*768 = 50,331,648

  auto gemm = [&](const float* X, const float* W, const float* bias,
                  const float* add, float* Y, int M, int N, int K,
                  int act = ACT_NONE, long wks = -1, long wns = 1,
                  int scatter = 0, int Wp = 0, int f = 0, int fy = 0,
                  int fx = 0, int wOut = 0) {
    if (wks < 0) wks = N;
    dim3 grid(N / 64, M / 32);
    k_gemm_wmma<<<grid, 256, 0, stream>>>(X, W, bias, add, Y, M, N, K, wks,
                                          wns, act, scatter, Wp, f, fy, fx,
                                          wOut);
  };

  auto run_block = [&](float* x, float* tmp, float* big, const Blk& B, int hh,
                       int ww, int L, int nh) {
    int ntok = hh * ww;
    int lnBlocks = (ntok * 32 + 255) / 256;
    k_layernorm<<<lnBlocks, 256, 0, stream>>>(x, B.n1_g, B.n1_b, tmp, ntok, L);
    gemm(tmp, B.qkv_w, B.qkv_b, nullptr, big, ntok, 3 * L, L);
    k_na2d<<<(ntok * nh + 255) / 256, 256, 0, stream>>>(big, tmp, hh, ww, nh);
    gemm(tmp, B.proj_w, B.proj_b, /*add=*/x, /*Y=*/x, ntok, L, L);
    k_layernorm<<<lnBlocks, 256, 0, stream>>>(x, B.n2_g, B.n2_b, tmp, ntok, L);
    gemm(tmp, B.fc1_w, B.fc1_b, nullptr, big, ntok, 4 * L, L, ACT_GELU);
    gemm(big, B.fc2_w, B.fc2_b, /*add=*/x, /*Y=*/x, ntok, L, 4 * L);
  };

  // ---- inputs -> x0 (32x32x16)
  k_concat<<<4, 256, 0, stream>>>(era5, rad, ang, stat, x0b);

  // ---- step 0: f=4, L=128, K=7, h=w=128, ff=2, prev = x0 (32x32x16)
  {
    int h = 128, w = 128, L = 128, ff = 2, f = 4, Hp = 32, Wp = 32, Cp = 16,
        nh = 8;
    float* x = R1;
    k_encode_pe<<<(h * w + 255) / 256, 256, 0, stream>>>(
        elev, modis, x0b, st2[0].enc_w, st2[0].enc_b, st2[0].conv_b, x, h, w,
        L, ff);
    for (int fy = 0; fy < f; ++fy)
      for (int fx = 0; fx < f; ++fx)
        gemm(x0b, st2[0].conv_w + (fy * f + fx), nullptr, /*add=*/x, x,
             Hp * Wp, L, Cp, ACT_NONE, (long)L * f * f, (long)f * f, 1, Wp, f,
             fy, fx, w);
    run_block(x, R2, R3, st2[0].blk[0], h, w, L, nh);
    run_block(x, R2, R3, st2[0].blk[1], h, w, L, nh);
  }

  // ---- step 1: f=2, L=192, K=7, h=w=256, ff=1, prev = R1 (128x128x128)
  {
    int h = 256, w = 256, L = 192, ff = 1, f = 2, Hp = 128, Wp = 128,
        Cp = 128, nh = 12;
    float* xprev = R1;
    float* x = R2;
    k_encode_pe<<<(h * w + 255) / 256, 256, 0, stream>>>(
        elev, modis, x0b, st2[1].enc_w, st2[1].enc_b, st2[1].conv_b, x, h, w,
        L, ff);
    for (int fy = 0; fy < f; ++fy)
      for (int fx = 0; fx < f; ++fx)
        gemm(xprev, st2[1].conv_w + (fy * f + fx), nullptr, /*add=*/x, x,
             Hp * Wp, L, Cp, ACT_NONE, (long)L * f * f, (long)f * f, 1, Wp, f,
             fy, fx, w);
    run_block(x, R1, R3, st2[1].blk[0], h, w, L, nh);
    run_block(x, R1, R3, st2[1].blk[1], h, w, L, nh);

    // ---- output head: relu(x @ w1 + b1) @ w2 + b2  -> (256*256, 8)
    gemm(x, out_w1, out_b1, nullptr, R3, 65536, 384, 192, ACT_RELU);
    k_head2<<<(65536 * 8 + 255) / 256, 256, 0, stream>>>(R3, out_w2, out_b2,
                                                         (float*)d_out, 65536);
  }
}